// TGCN_Model_8581344657590
// MI455X (gfx1250) — compile-verified
//
#include <hip/hip_runtime.h>
#include <math.h>

// MI455X / gfx1250 TGCN. H==0 collapses each cell (r-gate dead, Z*H term dead).
// fp32 WMMA (V_WMMA_F32_16X16X4_F32) for all dense GEMMs, fully unrolled K with
// fragment-packed zero-padded B. All strides are template constants so the hot
// loop is immediate-offset b64 load clauses + an unbroken WMMA chain, and the
// epilogue is 8 immediate-offset stores on the (common) full-tile path.

typedef __attribute__((ext_vector_type(2))) float v2f;
typedef __attribute__((ext_vector_type(8))) float v8f;

#define THREADS 256

// padded feature strides
#define FS1 112   // cell1 concat z|h features (100 used)
#define HS1 52    // h1 (50 used)
#define FS2 48    // cell2 concat z|h features (40 used)

__device__ __forceinline__ void atomic_add_f32(float* p, float v) {
  (void)__hip_atomic_fetch_add(p, v, __ATOMIC_RELAXED, __HIP_MEMORY_SCOPE_AGENT);
}

// Bp fragment-order index: pairs (k0,k0+1) per lane-half, per 4-k block
__device__ __forceinline__ int bpidx(int k, int nn, int Npad) {
  return (((k >> 2) * Npad + nn) * 2 + ((k >> 1) & 1)) * 2 + (k & 1);
}

// ---------------- degree / normalization ----------------
__global__ void k_init_deg(float* deg, int n) {
  int i = blockIdx.x * blockDim.x + threadIdx.x;
  if (i < n) deg[i] = 1.0f;  // self-loop weight
}

__global__ void k_edge_deg(const int* __restrict__ dst, const float* __restrict__ ew,
                           float* deg, int nE) {
  int e = blockIdx.x * blockDim.x + threadIdx.x;
  if (e < nE) atomic_add_f32(&deg[dst[e]], ew[e]);
}

__global__ void k_dinv(float* deg, int n) {
  int i = blockIdx.x * blockDim.x + threadIdx.x;
  if (i < n) { float d = deg[i]; deg[i] = d > 0.f ? rsqrtf(d) : 0.f; }
}

// ---------------- weight packing into fragment-ordered, zero-padded Bp ----------------
__global__ void k_pack(const float* __restrict__ wz1, const float* __restrict__ wh1,
                       const float* __restrict__ lzw1, const float* __restrict__ lhw1,
                       const float* __restrict__ wz2, const float* __restrict__ wh2,
                       const float* __restrict__ lzw2, const float* __restrict__ lhw2,
                       float* Wcat1p, float* BzT1p, float* BhT1p,
                       float* Wcat2p, float* BzT2p, float* BhT2p) {
  int t = blockIdx.x * blockDim.x + threadIdx.x;
  int stride = gridDim.x * blockDim.x;
  // Wcat1p: logical B[16x100] = [wz1|wh1], Kpad=16 Npad=112
  for (int idx = t; idx < 16 * 112; idx += stride) {
    int k = idx / 112, nn = idx % 112;
    float v = 0.f;
    if (nn < 100) v = (nn < 50) ? wz1[k * 50 + nn] : wh1[k * 50 + (nn - 50)];
    Wcat1p[bpidx(k, nn, 112)] = v;
  }
  // BzT1p/BhT1p: logical B[50x50] = lz_w/lh_w left half transposed, Kpad=52 Npad=64
  for (int idx = t; idx < 52 * 64; idx += stride) {
    int k = idx / 64, nn = idx % 64;
    int p = bpidx(k, nn, 64);
    bool in = (k < 50) && (nn < 50);
    BzT1p[p] = in ? lzw1[nn * 100 + k] : 0.f;
    BhT1p[p] = in ? lhw1[nn * 100 + k] : 0.f;
  }
  // Wcat2p: logical B[50x40] = [wz2|wh2], Kpad=52 Npad=48
  for (int idx = t; idx < 52 * 48; idx += stride) {
    int k = idx / 48, nn = idx % 48;
    float v = 0.f;
    if (k < 50 && nn < 40) v = (nn < 20) ? wz2[k * 20 + nn] : wh2[k * 20 + (nn - 20)];
    Wcat2p[bpidx(k, nn, 48)] = v;
  }
  // BzT2p/BhT2p: logical B[20x20], Kpad=20 Npad=32
  for (int idx = t; idx < 20 * 32; idx += stride) {
    int k = idx / 32, nn = idx % 32;
    int p = bpidx(k, nn, 32);
    bool in = (nn < 20);
    BzT2p[p] = in ? lzw2[nn * 40 + k] : 0.f;
    BhT2p[p] = in ? lhw2[nn * 40 + k] : 0.f;
  }
}

// ---------------- fp32 WMMA GEMM, K fully unrolled, const strides ----------------
template <int KPAD, int LDA, int NPAD, int LDC>
__global__ void k_wmma_gemm(const float* __restrict__ A,
                            const float* __restrict__ Bp,
                            float* __restrict__ C,
                            int M, int N) {
  constexpr int NKB = KPAD / 4;
  int lane  = threadIdx.x & 31;
  int wave  = threadIdx.x >> 5;
  int tileN = blockIdx.x * 16;
  int tileM = (blockIdx.y * 8 + wave) * 16;
  if (tileM >= M) return;  // wave-uniform: EXEC stays all-ones

  int mrow  = tileM + (lane & 15);
  int mload = mrow < M ? mrow : M - 1;  // clamp: masked rows never stored
  int ncol  = tileN + (lane & 15);
  int h     = lane >> 4;                // 0: K=0,1   1: K=2,3 (per k-block)

  const float* arow = A + (size_t)mload * LDA + 2 * h;
  const float* brow = Bp + (size_t)ncol * 4 + 2 * h;

  v2f af[NKB], bf[NKB];
#pragma unroll
  for (int kb = 0; kb < NKB; ++kb) {
    af[kb] = *(const v2f*)(arow + kb * 4);          // immediate offset kb*16B
    bf[kb] = *(const v2f*)(brow + kb * NPAD * 4);   // immediate offset kb*NPAD*16B
  }
  v8f acc = {};
#pragma unroll
  for (int kb = 0; kb < NKB; ++kb)
    acc = __builtin_amdgcn_wmma_f32_16x16x4_f32(false, af[kb], false, bf[kb],
                                                (short)0, acc, false, false);

  int rbase = tileM + (h << 3);  // lanes 16-31 hold M = v + 8
  if (tileM + 16 <= M && tileN + 16 <= N) {
    // full tile: 8 unpredicated stores at compile-time offsets v*LDC*4
    float* cbase = C + (size_t)rbase * LDC + ncol;
#pragma unroll
    for (int v = 0; v < 8; ++v) cbase[v * LDC] = acc[v];
  } else {
#pragma unroll
    for (int v = 0; v < 8; ++v) {
      int r = rbase + v;
      if (r < M && ncol < N) C[(size_t)r * LDC + ncol] = acc[v];
    }
  }
}

// ---------------- self-loop + bias init (zeroes pad columns) ----------------
__global__ void k_agg_init(const float* __restrict__ xw, const float* __restrict__ dinv,
                           const float* __restrict__ bz, const float* __restrict__ bh,
                           float* __restrict__ agg, int n, int FS, int Freal, int half) {
  int idx = blockIdx.x * blockDim.x + threadIdx.x;
  if (idx >= n * FS) return;
  int i = idx / FS, j = idx % FS;
  float v = 0.f;
  if (j < Freal) {
    float di = dinv[i];
    float b = (j < half) ? bz[j] : bh[j - half];
    v = di * di * xw[idx] + b;
  }
  agg[idx] = v;
}

// ---------------- edge aggregation: one wave32 per edge ----------------
__global__ void k_edge_agg(const int* __restrict__ src, const int* __restrict__ dst,
                           const float* __restrict__ ew, const float* __restrict__ dinv,
                           const float* __restrict__ xw, float* __restrict__ agg,
                           int nE, int FS, int Freal) {
  int gid  = blockIdx.x * blockDim.x + threadIdx.x;
  int e    = gid >> 5;
  int lane = gid & 31;
  if (e >= nE) return;
  int s = src[e], d = dst[e];
  float coef = dinv[s] * ew[e] * dinv[d];
  const float* xs = xw  + (size_t)s * FS;
  float*       ad = agg + (size_t)d * FS;
  for (int j = lane; j < Freal; j += 32)
    atomic_add_f32(ad + j, coef * xs[j]);
}

// ---------------- gate combine: h = relu((1-sigmoid(zp)) * tanh(hp)) ----------------
__global__ void k_combine(const float* __restrict__ pre, int ld, int half,
                          const float* __restrict__ zb, const float* __restrict__ hb,
                          float* __restrict__ out, int os, int n) {
  int idx = blockIdx.x * blockDim.x + threadIdx.x;
  if (idx >= n * os) return;
  int i = idx / os, j = idx % os;
  float v = 0.f;
  if (j < half) {
    float zp = pre[(size_t)i * ld + j] + zb[j];
    float hp = pre[(size_t)i * ld + half + j] + hb[j];
    float z  = 1.f / (1.f + __expf(-zp));
    float ht = tanhf(hp);
    float hv = (1.f - z) * ht;
    v = hv > 0.f ? hv : 0.f;
  }
  out[idx] = v;
}

// ---------------- final linear [n,20] @ [20,1] + b ----------------
__global__ void k_final(const float* __restrict__ h2, const float* __restrict__ lw,
                        const float* __restrict__ lb, float* __restrict__ out, int n) {
  int i = blockIdx.x * blockDim.x + threadIdx.x;
  if (i >= n) return;
  float s = lb[0];
#pragma unroll
  for (int j = 0; j < 20; ++j) s += h2[(size_t)i * 20 + j] * lw[j];
  out[i] = s;
}

extern "C" void kernel_launch(void* const* d_in, const int* in_sizes, int n_in,
                              void* d_out, int out_size, void* d_ws, size_t ws_size,
                              hipStream_t stream) {
  const float* x       = (const float*)d_in[0];
  const int*   ei      = (const int*)  d_in[1];
  const float* ew      = (const float*)d_in[2];
  const float* c1_wz   = (const float*)d_in[3];
  const float* c1_bz   = (const float*)d_in[4];
  const float* c1_lzw  = (const float*)d_in[5];
  const float* c1_lzb  = (const float*)d_in[6];
  // d_in[7..10] = c1 r-gate params: dead (H==0)
  const float* c1_wh   = (const float*)d_in[11];
  const float* c1_bh   = (const float*)d_in[12];
  const float* c1_lhw  = (const float*)d_in[13];
  const float* c1_lhb  = (const float*)d_in[14];
  const float* c2_wz   = (const float*)d_in[15];
  const float* c2_bz   = (const float*)d_in[16];
  const float* c2_lzw  = (const float*)d_in[17];
  const float* c2_lzb  = (const float*)d_in[18];
  // d_in[19..22] = c2 r-gate params: dead
  const float* c2_wh   = (const float*)d_in[23];
  const float* c2_bh   = (const float*)d_in[24];
  const float* c2_lhw  = (const float*)d_in[25];
  const float* c2_lhb  = (const float*)d_in[26];
  const float* lin_w   = (const float*)d_in[27];
  const float* lin_b   = (const float*)d_in[28];

  int n  = in_sizes[0] / 16;
  int nE = in_sizes[2];
  const int* src = ei;
  const int* dst = ei + nE;

  float* ws = (float*)d_ws;
  size_t o = 0;
  float* dinv   = ws + o; o += ((size_t)n + 127) & ~(size_t)127;
  float* bufA   = ws + o; o += (size_t)n * FS1;  // XW1 / gate-pre1 / XW2 / gate-pre2
  float* bufB   = ws + o; o += (size_t)n * FS1;  // AGG1 / h1 / AGG2 / h2
  float* Wcat1p = ws + o; o += 16 * 112;
  float* BzT1p  = ws + o; o += 52 * 64;
  float* BhT1p  = ws + o; o += 52 * 64;
  float* Wcat2p = ws + o; o += 52 * 48;
  float* BzT2p  = ws + o; o += 20 * 32;
  float* BhT2p  = ws + o; o += 20 * 32;

  auto cdiv = [](int a, int b) { return (a + b - 1) / b; };
  dim3 blk(THREADS);

  // normalization
  k_init_deg<<<cdiv(n, THREADS), blk, 0, stream>>>(dinv, n);
  k_edge_deg<<<cdiv(nE, THREADS), blk, 0, stream>>>(dst, ew, dinv, nE);
  k_dinv<<<cdiv(n, THREADS), blk, 0, stream>>>(dinv, n);
  k_pack<<<40, blk, 0, stream>>>(c1_wz, c1_wh, c1_lzw, c1_lhw,
                                 c2_wz, c2_wh, c2_lzw, c2_lhw,
                                 Wcat1p, BzT1p, BhT1p, Wcat2p, BzT2p, BhT2p);

  int gy = cdiv(n, 128);

  // ---- cell 1 (fin=16, fout=50, fused z|h width 100, stride FS1) ----
  k_wmma_gemm<16, 16, 112, FS1><<<dim3(7, gy), blk, 0, stream>>>(x, Wcat1p, bufA, n, 100);
  k_agg_init<<<cdiv(n * FS1, THREADS), blk, 0, stream>>>(bufA, dinv, c1_bz, c1_bh,
                                                         bufB, n, FS1, 100, 50);
  k_edge_agg<<<cdiv(nE * 32, THREADS), blk, 0, stream>>>(src, dst, ew, dinv,
                                                         bufA, bufB, nE, FS1, 100);
  k_wmma_gemm<52, FS1, 64, FS1><<<dim3(4, gy), blk, 0, stream>>>(bufB,      BzT1p, bufA,      n, 50);
  k_wmma_gemm<52, FS1, 64, FS1><<<dim3(4, gy), blk, 0, stream>>>(bufB + 50, BhT1p, bufA + 50, n, 50);
  k_combine<<<cdiv(n * HS1, THREADS), blk, 0, stream>>>(bufA, FS1, 50, c1_lzb, c1_lhb,
                                                        bufB, HS1, n);  // h1 stride 52, pads 0

  // ---- cell 2 (fin=50, fout=20, fused width 40, stride FS2) ----
  k_wmma_gemm<52, HS1, 48, FS2><<<dim3(3, gy), blk, 0, stream>>>(bufB, Wcat2p, bufA, n, 40);
  k_agg_init<<<cdiv(n * FS2, THREADS), blk, 0, stream>>>(bufA, dinv, c2_bz, c2_bh,
                                                         bufB, n, FS2, 40, 20);
  k_edge_agg<<<cdiv(nE * 32, THREADS), blk, 0, stream>>>(src, dst, ew, dinv,
                                                         bufA, bufB, nE, FS2, 40);
  k_wmma_gemm<20, FS2, 32, FS2><<<dim3(2, gy), blk, 0, stream>>>(bufB,      BzT2p, bufA,      n, 20);
  k_wmma_gemm<20, FS2, 32, FS2><<<dim3(2, gy), blk, 0, stream>>>(bufB + 20, BhT2p, bufA + 20, n, 20);
  k_combine<<<cdiv(n * 20, THREADS), blk, 0, stream>>>(bufA, FS2, 20, c2_lzb, c2_lhb,
                                                       bufB, 20, n);

  k_final<<<cdiv(n, THREADS), blk, 0, stream>>>(bufB, lin_w, lin_b, (float*)d_out, n);

  (void)n_in; (void)out_size; (void)ws_size;
}